// DistanceMax_47777216201318
// MI455X (gfx1250) — compile-verified
//
#include <hip/hip_runtime.h>

typedef float v2f __attribute__((ext_vector_type(2)));
typedef float v8f __attribute__((ext_vector_type(8)));

#define B_ROWS 8192
#define D_DIM  512
#define C_COLS 1000

#define TILE_M 128
#define TILE_N 128
#define TILE_K 32
#define XS_STRIDE 36    // 32 + 4 pad: conflict-free A reads, 16B-aligned
#define WS_STRIDE 144   // 128 + 16 pad: half-waves 32 banks apart on B reads

// ---------------------------------------------------------------------------
// Kernel 1: xsq[b] = sum_d X[b][d]^2   (one wave32 per row)
// ---------------------------------------------------------------------------
__global__ __launch_bounds__(256)
void xsq_kernel(const float* __restrict__ X, float* __restrict__ xsq) {
    const int lane = threadIdx.x & 31;
    const int row  = blockIdx.x * 8 + (threadIdx.x >> 5);
    const float* p = X + (size_t)row * D_DIM;
    float s = 0.0f;
    #pragma unroll
    for (int i = 0; i < D_DIM / 32; ++i) {
        float v = p[lane + i * 32];
        s += v * v;
    }
    #pragma unroll
    for (int off = 16; off > 0; off >>= 1)
        s += __shfl_xor(s, off, 32);
    if (lane == 0) xsq[row] = s;
}

// ---------------------------------------------------------------------------
// Kernel 2: wsq[c] = sum_d W[d][c]^2   (one thread per column, coalesced)
// ---------------------------------------------------------------------------
__global__ __launch_bounds__(256)
void wsq_kernel(const float* __restrict__ W, float* __restrict__ wsq) {
    const int c = blockIdx.x * 256 + threadIdx.x;
    if (c < C_COLS) {
        float s = 0.0f;
        for (int d = 0; d < D_DIM; ++d) {
            float v = W[(size_t)d * C_COLS + c];
            s += v * v;
        }
        wsq[c] = s;
    }
}

// ---------------------------------------------------------------------------
// Async global -> LDS staging of one K-chunk (per-wave ASYNCcnt tracked).
// Each thread issues 4x B128 for X tile and 4x B128 for W tile; data never
// touches VGPRs (CDNA5 GLOBAL_LOAD_ASYNC_TO_LDS path).
// ---------------------------------------------------------------------------
__device__ __forceinline__ void issue_chunk_async(
    const float* __restrict__ X, const float* __restrict__ W,
    int mBlock, int nBlock, int kb, int tid,
    float* XsBuf, float* WsBuf) {
    // X tile: 128 rows x 32 k = 1024 float4; 256 threads x 4
    #pragma unroll
    for (int it = 0; it < 4; ++it) {
        int q   = tid + it * 256;
        int row = q >> 3;             // 8 float4 per row
        int col = (q & 7) << 2;
        unsigned lds = (unsigned)(size_t)(&XsBuf[row * XS_STRIDE + col]);
        unsigned long long g =
            (unsigned long long)(size_t)(X + (size_t)(mBlock + row) * D_DIM + kb + col);
        asm volatile("global_load_async_to_lds_b128 %0, %1, off"
                     :: "v"(lds), "v"(g) : "memory");
    }
    // W tile: 32 k x 128 cols = 1024 float4; clamp OOB columns (safe: padded
    // N-columns only feed outputs with n >= C_COLS, which are never stored)
    #pragma unroll
    for (int it = 0; it < 4; ++it) {
        int q  = tid + it * 256;
        int d  = q >> 5;              // 32 float4 per k-row
        int c  = (q & 31) << 2;
        int gc = nBlock + c;
        if (gc > C_COLS - 4) gc = C_COLS - 4;   // clamp, stays 16B-aligned
        unsigned lds = (unsigned)(size_t)(&WsBuf[d * WS_STRIDE + c]);
        unsigned long long g =
            (unsigned long long)(size_t)(W + (size_t)(kb + d) * C_COLS + gc);
        asm volatile("global_load_async_to_lds_b128 %0, %1, off"
                     :: "v"(lds), "v"(g) : "memory");
    }
}

// ---------------------------------------------------------------------------
// Main kernel: cross = X @ W via V_WMMA_F32_16X16X4_F32, async double-buffered
// LDS staging, fused epilogue out[m][n] = -sqrt(max(xsq[m]+wsq[n]-2*cross,0)).
// Workgroup = 8 waves, tile 128(M) x 128(N); wave tile 64x32 (4x2 accums).
// ---------------------------------------------------------------------------
__global__ __launch_bounds__(256)
void gemm_dist_kernel(const float* __restrict__ X,
                      const float* __restrict__ W,
                      const float* __restrict__ xsq,
                      const float* __restrict__ wsq,
                      float* __restrict__ out) {
    __shared__ float Xs[2][TILE_M * XS_STRIDE];
    __shared__ float Ws[2][TILE_K * WS_STRIDE];

    const int tid  = threadIdx.x;
    const int lane = tid & 31;
    const int h    = lane >> 4;   // half-wave: 0 or 1
    const int l16  = lane & 15;
    const int wave = tid >> 5;
    const int wm   = wave & 1;    // 2 waves along M
    const int wn   = wave >> 1;   // 4 waves along N

    const int mBlock = blockIdx.y * TILE_M;
    const int nBlock = blockIdx.x * TILE_N;

    v8f acc[4][2];
    #pragma unroll
    for (int i = 0; i < 4; ++i)
        #pragma unroll
        for (int j = 0; j < 2; ++j)
            #pragma unroll
            for (int e = 0; e < 8; ++e)
                acc[i][j][e] = 0.0f;

    // Prologue: stage chunk 0 into buffer 0
    issue_chunk_async(X, W, mBlock, nBlock, 0, tid, Xs[0], Ws[0]);

    const int nChunks = D_DIM / TILE_K;    // 16
    for (int ch = 0; ch < nChunks; ++ch) {
        // Own async copies done -> barrier makes ALL waves' copies visible.
        // This same barrier also separates last iteration's compute on buffer
        // (ch+1)&1 from the issue below that refills it (double buffering),
        // so no second barrier per chunk is needed.
        asm volatile("s_wait_asynccnt 0" ::: "memory");
        __syncthreads();

        if (ch + 1 < nChunks)
            issue_chunk_async(X, W, mBlock, nBlock, (ch + 1) * TILE_K, tid,
                              Xs[(ch + 1) & 1], Ws[(ch + 1) & 1]);

        const float* Xc = Xs[ch & 1];
        const float* Wc = Ws[ch & 1];

        // --- 8 WMMA K-steps of 4 over this chunk ---
        #pragma unroll
        for (int kk = 0; kk < TILE_K; kk += 4) {
            // A fragments: component j, half h -> K = kk + j + 2h ; M = l16
            v2f a[4];
            #pragma unroll
            for (int i = 0; i < 4; ++i) {
                const float* p = &Xc[(wm * 64 + i * 16 + l16) * XS_STRIDE + kk + 2 * h];
                a[i][0] = p[0];
                a[i][1] = p[1];
            }
            // B fragments: component j, half h -> K = kk + j + 2h ; N = l16
            v2f b[2];
            #pragma unroll
            for (int j = 0; j < 2; ++j) {
                int col = wn * 32 + j * 16 + l16;
                b[j][0] = Wc[(kk + 2 * h + 0) * WS_STRIDE + col];
                b[j][1] = Wc[(kk + 2 * h + 1) * WS_STRIDE + col];
            }
            #pragma unroll
            for (int i = 0; i < 4; ++i)
                #pragma unroll
                for (int j = 0; j < 2; ++j)
                    acc[i][j] = __builtin_amdgcn_wmma_f32_16x16x4_f32(
                        false, a[i], false, b[j], (short)0, acc[i][j],
                        false, false);
        }
    }

    // --- fused epilogue: -sqrt(max(xsq + wsq - 2*cross, 0)) ---
    // Output is streamed (written once, never re-read): non-temporal stores
    // keep the 32.8 MB result from evicting the L2-resident X rows / W tiles.
    #pragma unroll
    for (int i = 0; i < 4; ++i) {
        const int mBase = mBlock + wm * 64 + i * 16;
        #pragma unroll
        for (int j = 0; j < 2; ++j) {
            const int n = nBlock + wn * 32 + j * 16 + l16;
            const bool nOK = (n < C_COLS);
            const float wq  = nOK ? wsq[n] : 0.0f;
            #pragma unroll
            for (int e = 0; e < 8; ++e) {
                const int m = mBase + e + 8 * h;     // C/D layout: M = e + 8h
                const float cross = acc[i][j][e];
                const float xq = xsq[m];
                const float sq = fmaxf(xq + wq - 2.0f * cross, 0.0f);
                if (nOK)
                    __builtin_nontemporal_store(-__builtin_sqrtf(sq),
                                                &out[(size_t)m * C_COLS + n]);
            }
        }
    }
}

// ---------------------------------------------------------------------------
extern "C" void kernel_launch(void* const* d_in, const int* in_sizes, int n_in,
                              void* d_out, int out_size, void* d_ws, size_t ws_size,
                              hipStream_t stream) {
    const float* X = (const float*)d_in[0];   // [8192, 512]
    const float* W = (const float*)d_in[1];   // [512, 1000]
    float* out = (float*)d_out;               // [8192, 1000]

    float* xsq = (float*)d_ws;                // 8192 floats
    float* wsq = xsq + B_ROWS;                // 1000 floats

    xsq_kernel<<<B_ROWS / 8, 256, 0, stream>>>(X, xsq);
    wsq_kernel<<<(C_COLS + 255) / 256, 256, 0, stream>>>(W, wsq);

    dim3 grid(8 /* ceil(1000/128) */, B_ROWS / TILE_M);
    gemm_dist_kernel<<<grid, 256, 0, stream>>>(X, W, xsq, wsq, out);
}